// HyperConnectionModule_40046275068100
// MI455X (gfx1250) — compile-verified
//
#include <hip/hip_runtime.h>
#include <math.h>

typedef __attribute__((ext_vector_type(2))) float v2f;
typedef __attribute__((ext_vector_type(4))) float v4f;
typedef __attribute__((ext_vector_type(8))) float v8f;

#define EPS_ 1e-6f
#define SINK_ITERS 20
#define N_ 4
#define D_ 4096
#define ND_ (N_ * D_)        // 16384
#define M_ 24                // n*n + 2n
#define ROWS_PER_BLK 16
#define THREADS 256

// ---- WMMA f32 16x16x4 wrapper (full fp32 precision matrix op on CDNA5) ----
#ifdef __has_builtin
#if __has_builtin(__builtin_amdgcn_wmma_f32_16x16x4_f32)
#define HAVE_WMMA_F32X4 1
#endif
#endif

__device__ __forceinline__ v8f wmma_f32_16x16x4(v2f a, v2f b, v8f c) {
#ifdef HAVE_WMMA_F32X4
  // (neg_a, A, neg_b, B, c_mod, C, reuse_a, reuse_b)
  return __builtin_amdgcn_wmma_f32_16x16x4_f32(false, a, false, b, (short)0, c,
                                               false, false);
#else
  asm volatile("v_wmma_f32_16x16x4_f32 %0, %1, %2, %0"
               : "+v"(c)
               : "v"(a), "v"(b));
  return c;
#endif
}

__global__ __launch_bounds__(THREADS) void hyperconn_fused(
    const float* __restrict__ R,      // residuals (bs, 4, 4096)
    const float* __restrict__ W,      // w_proj    (24, 16384)
    const float* __restrict__ bias,   // (24,)
    const float* __restrict__ apre_p, const float* __restrict__ apost_p,
    const float* __restrict__ ares_p,
    float* __restrict__ O)            // (bs, 4, 4096)
{
  __shared__ float s_cpart[8 * 8 * 32];          // per-wave C fragments
  __shared__ float s_ss[4 * 32];                 // sum-of-squares partials
  __shared__ float s_proj[ROWS_PER_BLK * 32];    // proj (16 rows x 32 cols, 24 valid)
  __shared__ float s_rr[ROWS_PER_BLK];           // rsqrt(mean(x^2)+eps)
  __shared__ float s_coef[ROWS_PER_BLK * 24];    // h_pre|h_post|h_res per row

  const int tid    = threadIdx.x;
  const int lane   = tid & 31;
  const int wave   = tid >> 5;       // 0..7
  const int nt     = wave & 1;       // N-tile: global cols [nt*16, nt*16+16)
  const int ks     = wave >> 1;      // K-slice: [ks*4096, ks*4096+4096)
  const int laneM  = lane & 15;      // matrix row (A) / col (B) for this lane
  const int laneHi = lane >> 4;      // 0: K=k,k+1   1: K=k+2,k+3
  const long b0    = (long)blockIdx.x * ROWS_PER_BLK;

  // ===================== Phase A: proj GEMM + sum(x^2) =====================
  // A layout (16x4 f32): lanes 0-15 -> {K=0,K=1}, lanes 16-31 -> {K=2,K=3}.
  // B layout (4x16 f32): mirrored, lane = N.
  const int col  = nt * 16 + laneM;
  const int wcol = (col < M_) ? col : (M_ - 1);  // clamp; padded cols discarded
  const float* aptr = R + (size_t)(b0 + laneM) * ND_ + (size_t)ks * 4096 + 2 * laneHi;
  const float* bptr = W + (size_t)wcol * ND_         + (size_t)ks * 4096 + 2 * laneHi;

  v8f c0 = {0, 0, 0, 0, 0, 0, 0, 0};
  v8f c1 = {0, 0, 0, 0, 0, 0, 0, 0};
  float ss = 0.0f;
#pragma unroll 4
  for (int k = 0; k < 4096; k += 8) {
    v2f a0 = *(const v2f*)(aptr + k);
    v2f b0v = *(const v2f*)(bptr + k);
    v2f a1 = *(const v2f*)(aptr + k + 4);
    v2f b1v = *(const v2f*)(bptr + k + 4);
    c0 = wmma_f32_16x16x4(a0, b0v, c0);
    c1 = wmma_f32_16x16x4(a1, b1v, c1);
    ss += a0.x * a0.x + a0.y * a0.y + a1.x * a1.x + a1.y * a1.y;  // co-exec VALU
  }
  v8f cacc = c0 + c1;
#pragma unroll
  for (int v = 0; v < 8; ++v) s_cpart[(wave * 8 + v) * 32 + lane] = cacc[v];
  if (nt == 0) s_ss[ks * 32 + lane] = ss;  // nt==1 duplicates the same X rows
  __syncthreads();

  // Reduce K-slice partials. C layout: VGPR v, lane<16 -> (M=v,N=lane),
  // lane>=16 -> (M=v+8, N=lane-16).
  for (int idx = tid; idx < 512; idx += THREADS) {
    const int tnt = idx >> 8;
    const int M   = (idx >> 4) & 15;
    const int N   = idx & 15;
    const int v   = M & 7;
    const int l   = N + ((M >> 3) << 4);
    float s = 0.0f;
#pragma unroll
    for (int kk = 0; kk < 4; ++kk) s += s_cpart[((kk * 2 + tnt) * 8 + v) * 32 + l];
    s_proj[M * 32 + tnt * 16 + N] = s;
  }
  if (tid < ROWS_PER_BLK) {
    float t = 0.0f;
#pragma unroll
    for (int kk = 0; kk < 4; ++kk)
      t += s_ss[kk * 32 + tid] + s_ss[kk * 32 + 16 + tid];
    s_rr[tid] = rsqrtf(t * (1.0f / (float)ND_) + EPS_);
  }
  __syncthreads();

  // ============== Phase B: sigmoid gates + Sinkhorn (per row) ==============
  if (tid < ROWS_PER_BLK) {
    const float rv    = s_rr[tid];
    const float apre  = apre_p[0];
    const float apost = apost_p[0];
    const float ares  = ares_p[0];
    float g[24];
#pragma unroll
    for (int m = 0; m < 24; ++m) {
      const float al = (m < 4) ? apre : ((m < 8) ? apost : ares);
      const float z  = al * (s_proj[tid * 32 + m] * rv) + bias[m];
      g[m] = 1.0f / (1.0f + __expf(-z));
    }
    float h[16];
#pragma unroll
    for (int i = 0; i < 16; ++i) h[i] = g[8 + i];
    for (int it = 0; it < SINK_ITERS; ++it) {
#pragma unroll
      for (int i = 0; i < 4; ++i) {
        const float inv =
            1.0f / (h[i * 4] + h[i * 4 + 1] + h[i * 4 + 2] + h[i * 4 + 3] + EPS_);
        h[i * 4] *= inv; h[i * 4 + 1] *= inv; h[i * 4 + 2] *= inv; h[i * 4 + 3] *= inv;
      }
#pragma unroll
      for (int j = 0; j < 4; ++j) {
        const float inv = 1.0f / (h[j] + h[4 + j] + h[8 + j] + h[12 + j] + EPS_);
        h[j] *= inv; h[4 + j] *= inv; h[8 + j] *= inv; h[12 + j] *= inv;
      }
    }
#pragma unroll
    for (int m = 0; m < 8; ++m) s_coef[tid * 24 + m] = g[m];
#pragma unroll
    for (int i = 0; i < 16; ++i) s_coef[tid * 24 + 8 + i] = h[i];
  }
  __syncthreads();

  // ====== Phase C: recombine (reads hit L2: rows were just streamed) =======
  for (int bl = 0; bl < ROWS_PER_BLK; ++bl) {
    const float* cf = &s_coef[bl * 24];
    float hp[4], hq[4], hr[16];
#pragma unroll
    for (int j = 0; j < 4; ++j) { hp[j] = cf[j]; hq[j] = cf[4 + j]; }
#pragma unroll
    for (int i = 0; i < 16; ++i) hr[i] = cf[8 + i];
    const size_t base = (size_t)(b0 + bl) * ND_;
#pragma unroll
    for (int it = 0; it < D_ / (THREADS * 4); ++it) {  // 4 iterations
      const int d = (it * THREADS + tid) * 4;
      const v4f r0 = *(const v4f*)(R + base + 0 * D_ + d);
      const v4f r1 = *(const v4f*)(R + base + 1 * D_ + d);
      const v4f r2 = *(const v4f*)(R + base + 2 * D_ + d);
      const v4f r3 = *(const v4f*)(R + base + 3 * D_ + d);
      const v4f xin = hp[0] * r0 + hp[1] * r1 + hp[2] * r2 + hp[3] * r3;
#pragma unroll
      for (int i = 0; i < 4; ++i) {
        v4f o = hr[i * 4 + 0] * r0 + hr[i * 4 + 1] * r1 + hr[i * 4 + 2] * r2 +
                hr[i * 4 + 3] * r3 + hq[i] * xin;
        *(v4f*)(O + base + i * D_ + d) = o;
      }
    }
  }
}

extern "C" void kernel_launch(void* const* d_in, const int* in_sizes, int n_in,
                              void* d_out, int out_size, void* d_ws, size_t ws_size,
                              hipStream_t stream) {
  (void)n_in; (void)out_size; (void)d_ws; (void)ws_size;
  const float* R     = (const float*)d_in[0];
  const float* W     = (const float*)d_in[1];
  const float* bias  = (const float*)d_in[2];
  const float* apre  = (const float*)d_in[3];
  const float* apost = (const float*)d_in[4];
  const float* ares  = (const float*)d_in[5];
  float* O = (float*)d_out;

  const int bs = in_sizes[0] / ND_;           // 4096
  const int nblocks = bs / ROWS_PER_BLK;      // 256
  hyperconn_fused<<<dim3(nblocks), dim3(THREADS), 0, stream>>>(
      R, W, bias, apre, apost, ares, O);
}